// KNN_8031588843521
// MI455X (gfx1250) — compile-verified
//
#include <hip/hip_runtime.h>
#include <math.h>

typedef __attribute__((ext_vector_type(2))) float v2f;
typedef __attribute__((ext_vector_type(8))) float v8f;

#define KNN_K 16
#define NREF  8192
#define NQ    4096
#define NB    4
#define WPB   8   // waves per block

// Branchless sorted-ascending insert. Pure per-slot selects: no dynamic
// indexing, no position integer -> avoids the cndmask select-tree lowering.
__device__ __forceinline__ void insert16(float (&best)[KNN_K], int (&bidx)[KNN_K],
                                         float s, int id) {
    bool c[KNN_K];
#pragma unroll
    for (int j = 0; j < KNN_K; ++j) c[j] = s < best[j];
#pragma unroll
    for (int j = KNN_K - 1; j > 0; --j) {           // descending: reads see old values
        best[j] = c[j - 1] ? best[j - 1] : (c[j] ? s : best[j]);
        bidx[j] = c[j - 1] ? bidx[j - 1] : (c[j] ? id : bidx[j]);
    }
    best[0] = c[0] ? s : best[0];
    bidx[0] = c[0] ? id : bidx[0];
}

// One wave = 16 queries. Per 16-ref chunk one V_WMMA_F32_16X16X4_F32 gives
//   s[r][q] = ||r||^2 - 2 r.q   (fp32-exact; ranking-equivalent to sq dist)
// A row (refs,  M=16,K=4) = (-2rx, -2ry, -2rz, r2)
// B col (query, K=4,N=16) = ( qx,   qy,   qz,  1)
__global__ __launch_bounds__(256) void knn_wmma_kernel(
    const float* __restrict__ ref, const float* __restrict__ query,
    float* __restrict__ out_dist, int* __restrict__ out_idx)
{
    __shared__ float sS[WPB][32][KNN_K + 1];
    __shared__ int   sI[WPB][32][KNN_K + 1];

    const int lane = threadIdx.x & 31;
    const int wv   = threadIdx.x >> 5;
    const int gw   = blockIdx.x * WPB + wv;
    const int tilesPerBatch = NQ / 16;               // 256
    const int b     = gw / tilesPerBatch;
    const int qtile = (gw % tilesPerBatch) * 16;

    const int  n  = lane & 15;
    const bool hi = lane >= 16;

    // ---- B operand: query column n (loaded once) ----
    const float* qp = query + ((size_t)b * NQ + qtile + n) * 3;
    const float qx = qp[0], qy = qp[1], qz = qp[2];
    const float q2 = qx * qx + qy * qy + qz * qz;
    v2f Bv;
    Bv.x = hi ? qz : qx;
    Bv.y = hi ? 1.0f : qy;

    float best[KNN_K];
    int   bidx[KNN_K];
#pragma unroll
    for (int j = 0; j < KNN_K; ++j) { best[j] = __builtin_inff(); bidx[j] = 0; }

    const float* rbase = ref + (size_t)b * NREF * 3;

#pragma unroll 2
    for (int c0 = 0; c0 < NREF; c0 += 16) {
        // A operand: ref row m = c0 + (lane&15); load all 3 comps in every lane
        const float* rp = rbase + (size_t)(c0 + n) * 3;
        const float rx = rp[0], ry = rp[1], rz = rp[2];
        const float r2 = fmaf(rx, rx, fmaf(ry, ry, rz * rz));
        v2f Av;
        Av.x = -2.0f * (hi ? rz : rx);
        Av.y = hi ? r2 : (-2.0f * ry);

        v8f acc = {};
        acc = __builtin_amdgcn_wmma_f32_16x16x4_f32(
            false, Av, false, Bv, (short)0, acc, false, false);

        const int idbase = c0 + (hi ? 8 : 0);
#pragma unroll
        for (int v = 0; v < 8; ++v) {
            const float s = acc[v];
            if (s < best[KNN_K - 1]) {               // rare wave-level branch
                insert16(best, bidx, s, idbase + v);
            }
        }
    }

    // ---- spill per-lane lists to LDS (+inf sentinel for merge) ----
#pragma unroll
    for (int j = 0; j < KNN_K; ++j) {
        sS[wv][lane][j] = best[j];
        sI[wv][lane][j] = bidx[j];
    }
    sS[wv][lane][KNN_K] = __builtin_inff();
    sI[wv][lane][KNN_K] = 0;
    __syncthreads();

    // ---- lanes 0..15: two-pointer merge with lane+16 (same query) ----
    if (!hi) {
        int ia = 0, ib = 0;
        const int    q     = qtile + n;
        const size_t obase = ((size_t)b * NQ + q) * KNN_K;
        float* dout = out_dist + obase;
        int*   iout = out_idx + obase;
#pragma unroll
        for (int j = 0; j < KNN_K; ++j) {
            const float sa = sS[wv][lane][ia];
            const float sb = sS[wv][lane + 16][ib];
            const bool  ta = sa <= sb;
            const float s  = ta ? sa : sb;
            const int   id = ta ? sI[wv][lane][ia] : sI[wv][lane + 16][ib];
            ia += ta ? 1 : 0;
            ib += ta ? 0 : 1;
            const float d2 = q2 + s;
            dout[j] = sqrtf(d2 > 0.0f ? d2 : 0.0f);
            iout[j] = id;
        }
    }
}

extern "C" void kernel_launch(void* const* d_in, const int* in_sizes, int n_in,
                              void* d_out, int out_size, void* d_ws, size_t ws_size,
                              hipStream_t stream) {
    const float* ref   = (const float*)d_in[0];   // [B, 8192, 3] f32
    const float* query = (const float*)d_in[1];   // [B, 4096, 3] f32

    float* dist = (float*)d_out;                                   // [B, Nq, 16] f32
    int*   idx  = (int*)((float*)d_out + (size_t)NB * NQ * KNN_K); // [B, Nq, 16] i32

    const int waves  = NB * (NQ / 16);        // 1024 waves
    const int blocks = waves / WPB;           // 128 blocks x 256 threads
    knn_wmma_kernel<<<blocks, 256, 0, stream>>>(ref, query, dist, idx);
}